// KNRM_76020921139730
// MI455X (gfx1250) — compile-verified
//
#include <hip/hip_runtime.h>
#include <math.h>

typedef float v2f __attribute__((ext_vector_type(2)));
typedef float v8f __attribute__((ext_vector_type(8)));

#define NKERN 11
#define BB 256
#define QQ 32
#define DD 256
#define EE 300
#define QSTRIDE 308   // LDS row stride for q tile (8B aligned, bank-conflict free)
#define CSTRIDE 260   // LDS row stride for cosine tile

__device__ __constant__ float KMU[NKERN] = {
    1.0f, 0.9f, 0.7f, 0.5f, 0.3f, 0.1f, -0.1f, -0.3f, -0.5f, -0.7f, -0.9f};
// coeff = -1/(2*sigma^2): sigma0 = 1e-4 -> -5e7 ; rest sigma = 0.1 -> -50
__device__ __constant__ float KCOEF[NKERN] = {
    -5.0e7f, -50.f, -50.f, -50.f, -50.f, -50.f, -50.f, -50.f, -50.f, -50.f, -50.f};

__global__ __launch_bounds__(256) void knrm_fused(
    const float* __restrict__ qemb,   // [B,Q,E]
    const float* __restrict__ demb,   // [B,D,E]
    const int*   __restrict__ qtok,   // [B,Q]
    const int*   __restrict__ dtok,   // [B,D]
    const float* __restrict__ wdense, // [1,NKERN]
    float*       __restrict__ out)    // [B]
{
    __shared__ __align__(16) float sh_buf[QQ * QSTRIDE]; // q tile, then cosine tile
    __shared__ float sh_qinv[QQ];
    __shared__ float sh_qm[QQ];
    __shared__ float sh_dinv[DD];
    __shared__ float sh_dm[DD];
    __shared__ float sh_acc[QQ * NKERN];
    __shared__ float sh_w[NKERN];

    const int b    = blockIdx.x;
    const int tid  = threadIdx.x;
    const int lane = tid & 31;
    const int wave = tid >> 5;

    // ---- stage query tile (raw, normalization applied later to C tile) ----
    const float* qb = qemb + (size_t)b * QQ * EE;
    for (int i = tid; i < QQ * EE; i += 256) {
        int r = i / EE;
        int e = i - r * EE;
        sh_buf[r * QSTRIDE + e] = qb[(size_t)r * EE + e];
    }
    if (tid < QQ)    sh_qm[tid] = (qtok[b * QQ + tid] > 1) ? 1.0f : 0.0f;
    sh_dm[tid] = (dtok[b * DD + tid] > 1) ? 1.0f : 0.0f;
    if (tid < NKERN) sh_w[tid] = wdense[tid];
    __syncthreads();

    // ---- GEMM: wave -> q-tile (wave&1), d-tiles {dbase, +4, +8, +12} ----
    const int qt    = wave & 1;
    const int dbase = wave >> 1;
    {
        const int n  = lane & 15;            // A: M index / B: N index
        const int ak = (lane >> 4) << 1;     // K sub-offset {0,2}
        const float* aptr = sh_buf + (qt * 16 + n) * QSTRIDE + ak;
        const float* bptr = demb + ((size_t)b * DD + (size_t)(dbase * 16 + n)) * EE + ak;

        v8f acc[4] = {{0,0,0,0,0,0,0,0}, {0,0,0,0,0,0,0,0},
                      {0,0,0,0,0,0,0,0}, {0,0,0,0,0,0,0,0}};
        float qs = 0.0f;
        float bs[4] = {0.0f, 0.0f, 0.0f, 0.0f};

        for (int k0 = 0; k0 < EE; k0 += 4) {
            v2f a = *(const v2f*)(aptr + k0);
            qs += a.x * a.x + a.y * a.y;
            #pragma unroll
            for (int j = 0; j < 4; ++j) {
                v2f bb = *(const v2f*)(bptr + (size_t)j * 64 * EE + k0);
                bs[j] += bb.x * bb.x + bb.y * bb.y;
                acc[j] = __builtin_amdgcn_wmma_f32_16x16x4_f32(
                    /*neg_a=*/false, a, /*neg_b=*/false, bb,
                    /*c_mod=*/(short)0, acc[j],
                    /*reuse_a=*/false, /*reuse_b=*/false);
            }
        }

        // ---- finish row norms: lane l covers k%4 in {ak,ak+1}; xor16 completes ----
        qs += __shfl_xor(qs, 16, 32);
        if (dbase == 0 && lane < 16)
            sh_qinv[qt * 16 + lane] = 1.0f / (sqrtf(qs) + 1e-13f);
        #pragma unroll
        for (int j = 0; j < 4; ++j) {
            bs[j] += __shfl_xor(bs[j], 16, 32);
            if (qt == 0 && lane < 16)
                sh_dinv[dbase * 16 + j * 64 + lane] = 1.0f / (sqrtf(bs[j]) + 1e-13f);
        }

        __syncthreads(); // all q-tile reads + inv-norm writes done

        // ---- scale C tile -> masked cosine, park in LDS (reusing sh_buf) ----
        const int mlo = (lane >> 4) << 3; // 0 or 8
        #pragma unroll
        for (int j = 0; j < 4; ++j) {
            int d = dbase * 16 + j * 64 + n;
            float dscale = sh_dinv[d] * sh_dm[d];
            #pragma unroll
            for (int r = 0; r < 8; ++r) {
                int qrow = qt * 16 + mlo + r;
                float c = acc[j][r] * sh_qinv[qrow] * sh_qm[qrow] * dscale;
                sh_buf[qrow * CSTRIDE + d] = c;
            }
        }
    }
    __syncthreads();

    // ---- gaussian kernel pooling: thread (q = tid>>3, d0 = tid&7) ----
    {
        const int   qrow = tid >> 3;
        const int   d0   = tid & 7;
        const float qmv  = sh_qm[qrow];
        float s[NKERN];
        #pragma unroll
        for (int k = 0; k < NKERN; ++k) s[k] = 0.0f;

        for (int d = d0; d < DD; d += 8) {
            float c = sh_buf[qrow * CSTRIDE + d];
            float m = qmv * sh_dm[d];
            #pragma unroll
            for (int k = 0; k < NKERN; ++k) {
                float diff = c - KMU[k];
                s[k] += m * __expf(KCOEF[k] * diff * diff);
            }
        }
        #pragma unroll
        for (int k = 0; k < NKERN; ++k) {
            s[k] += __shfl_xor(s[k], 1, 32);
            s[k] += __shfl_xor(s[k], 2, 32);
            s[k] += __shfl_xor(s[k], 4, 32);
        }
        if (d0 == 0) {
            #pragma unroll
            for (int k = 0; k < NKERN; ++k) sh_acc[qrow * NKERN + k] = s[k];
        }
    }
    __syncthreads();

    // ---- log / mask / sum over q, dot with w (wave 0) ----
    if (tid < QQ) {
        float qmv = sh_qm[tid];
        float sq = 0.0f;
        #pragma unroll
        for (int k = 0; k < NKERN; ++k) {
            float v = logf(fmaxf(sh_acc[tid * NKERN + k], 1e-10f)) * 0.01f * qmv;
            sq += v * sh_w[k];
        }
        sq += __shfl_xor(sq, 1, 32);
        sq += __shfl_xor(sq, 2, 32);
        sq += __shfl_xor(sq, 4, 32);
        sq += __shfl_xor(sq, 8, 32);
        sq += __shfl_xor(sq, 16, 32);
        if (tid == 0) out[b] = sq;
    }
}

extern "C" void kernel_launch(void* const* d_in, const int* in_sizes, int n_in,
                              void* d_out, int out_size, void* d_ws, size_t ws_size,
                              hipStream_t stream) {
    (void)in_sizes; (void)n_in; (void)out_size; (void)d_ws; (void)ws_size;
    const float* qemb   = (const float*)d_in[0];
    const float* demb   = (const float*)d_in[1];
    const int*   qtok   = (const int*)d_in[2];
    const int*   dtok   = (const int*)d_in[3];
    const float* wdense = (const float*)d_in[4];
    float* out = (float*)d_out;

    knrm_fused<<<BB, 256, 0, stream>>>(qemb, demb, qtok, dtok, wdense, out);
}